// GINELayer_30116310679888
// MI455X (gfx1250) — compile-verified
//
#include <hip/hip_runtime.h>

typedef float v2f __attribute__((ext_vector_type(2)));
typedef float v8f __attribute__((ext_vector_type(8)));

#define N_NODES 100000
#define N_EDGES 1600000
#define IN_F 128
#define OUT_F 128
#define EDGE_F 16

// ---------------------------------------------------------------------------
// Zero the aggregation workspace (graph-capture safe; vectorized stores).
// ---------------------------------------------------------------------------
__global__ __launch_bounds__(256) void gine_zero_kernel(float4* __restrict__ p, int n4) {
    int i = blockIdx.x * blockDim.x + threadIdx.x;
    if (i < n4) {
        float4 z;
        z.x = 0.f; z.y = 0.f; z.z = 0.f; z.w = 0.f;
        p[i] = z;
    }
}

// ---------------------------------------------------------------------------
// Edge pass: msg = edge_attr @ We + be + H[src]; atomic scatter-add by dst.
// One wave handles a 16-edge x 128-feature tile via 32x v_wmma_f32_16x16x4_f32.
// ---------------------------------------------------------------------------
__global__ __launch_bounds__(256) void gine_edge_kernel(
    const float* __restrict__ H,
    const long long* __restrict__ ei,      // [2, E] int64: row0 = src, row1 = dst
    const float* __restrict__ EA,          // [E, 16]
    const float* __restrict__ We,          // [16, 128]
    const float* __restrict__ be,          // [128]
    float* __restrict__ agg)               // [N, 128]
{
    const int lane = threadIdx.x & 31;
    const int wave = threadIdx.x >> 5;
    const int e0   = (blockIdx.x * 8 + wave) * 16;   // 16 edges per wave
    const int nlo  = lane & 15;
    const int hi   = lane >> 4;

    // A-matrix: 16x16 edge_attr tile as four 16x4 fp32 chunks.
    // Lane layout (ISA 7.12.2): lanes 0-15 -> K=k0,k0+1 ; lanes 16-31 -> K=k0+2,k0+3.
    v2f a[4];
#pragma unroll
    for (int k = 0; k < 4; ++k) {
        const float* p = EA + (size_t)(e0 + nlo) * EDGE_F + k * 4 + 2 * hi;
        a[k].x = p[0];
        a[k].y = p[1];
    }

    v8f acc[8] = {};  // 8 N-tiles of 16 columns each -> full 128 output features

#pragma unroll
    for (int k = 0; k < 4; ++k) {
        const int kr = k * 4 + 2 * hi;
#pragma unroll
        for (int nt = 0; nt < 8; ++nt) {
            v2f b;
            b.x = We[(size_t)kr * IN_F + nt * 16 + nlo];
            b.y = We[(size_t)(kr + 1) * IN_F + nt * 16 + nlo];
            acc[nt] = __builtin_amdgcn_wmma_f32_16x16x4_f32(
                false, a[k], false, b, (short)0, acc[nt], false, false);
        }
    }

    // Rows owned by this lane in C layout: m = i + 8*hi, i = 0..7.
    int srcOff[8], dstOff[8];
#pragma unroll
    for (int i = 0; i < 8; ++i) {
        const int m = 8 * hi + i;
        srcOff[i] = (int)ei[(size_t)e0 + m] * IN_F;
        dstOff[i] = (int)ei[(size_t)N_EDGES + e0 + m] * IN_F;
    }

    float bev[8];
#pragma unroll
    for (int nt = 0; nt < 8; ++nt) bev[nt] = be[nt * 16 + nlo];

#pragma unroll
    for (int nt = 0; nt < 8; ++nt) {
        const int n = nt * 16 + nlo;
#pragma unroll
        for (int i = 0; i < 8; ++i) {
            const float v = acc[nt][i] + bev[nt] + H[srcOff[i] + n];
            unsafeAtomicAdd(&agg[dstOff[i] + n], v);   // global_atomic_add_f32
        }
    }
}

// ---------------------------------------------------------------------------
// Node pass: out = relu(agg @ W1 + b1) @ W2 + b2, fused.
// One wave (32-thread block) per 16-node tile; LDS transposes h from the
// WMMA C-layout back to the A-layout between the two GEMMs.
// ---------------------------------------------------------------------------
__global__ __launch_bounds__(32) void gine_node_kernel(
    const float* __restrict__ agg,   // [N, 128]
    const float* __restrict__ W1,    // [128, 128]
    const float* __restrict__ b1,    // [128]
    const float* __restrict__ W2,    // [128, 128]
    const float* __restrict__ b2,    // [128]
    float* __restrict__ out)         // [N, 128]
{
    __shared__ float lds[16][IN_F + 4];   // +4 floats padding: avoid bank conflicts

    const int lane  = threadIdx.x;
    const int node0 = blockIdx.x * 16;
    const int nlo   = lane & 15;
    const int hi    = lane >> 4;

    // ---- GEMM1: agg(16x128) @ W1(128x128) ----
    v8f acc[8] = {};
#pragma unroll 4
    for (int k0 = 0; k0 < IN_F; k0 += 4) {
        v2f a;
        const float* p = agg + (size_t)(node0 + nlo) * IN_F + k0 + 2 * hi;
        a.x = p[0];
        a.y = p[1];
        const int kr = k0 + 2 * hi;
#pragma unroll
        for (int nt = 0; nt < 8; ++nt) {
            v2f b;
            b.x = W1[(size_t)kr * OUT_F + nt * 16 + nlo];
            b.y = W1[(size_t)(kr + 1) * OUT_F + nt * 16 + nlo];
            acc[nt] = __builtin_amdgcn_wmma_f32_16x16x4_f32(
                false, a, false, b, (short)0, acc[nt], false, false);
        }
    }

    // bias + ReLU, stash h tile into LDS (C-layout positions -> row-major tile)
#pragma unroll
    for (int nt = 0; nt < 8; ++nt) {
        const int n = nt * 16 + nlo;
        const float bias = b1[n];
#pragma unroll
        for (int i = 0; i < 8; ++i) {
            const float v = acc[nt][i] + bias;
            lds[i + 8 * hi][n] = v > 0.f ? v : 0.f;
        }
    }
    __syncthreads();

    // ---- GEMM2: h(16x128) @ W2(128x128) ----
    v8f acc2[8] = {};
#pragma unroll 4
    for (int k0 = 0; k0 < OUT_F; k0 += 4) {
        v2f a;
        a.x = lds[nlo][k0 + 2 * hi];
        a.y = lds[nlo][k0 + 2 * hi + 1];
        const int kr = k0 + 2 * hi;
#pragma unroll
        for (int nt = 0; nt < 8; ++nt) {
            v2f b;
            b.x = W2[(size_t)kr * OUT_F + nt * 16 + nlo];
            b.y = W2[(size_t)(kr + 1) * OUT_F + nt * 16 + nlo];
            acc2[nt] = __builtin_amdgcn_wmma_f32_16x16x4_f32(
                false, a, false, b, (short)0, acc2[nt], false, false);
        }
    }

    // bias + store
#pragma unroll
    for (int nt = 0; nt < 8; ++nt) {
        const int n = nt * 16 + nlo;
        const float bias = b2[n];
#pragma unroll
        for (int i = 0; i < 8; ++i) {
            out[(size_t)(node0 + i + 8 * hi) * OUT_F + n] = acc2[nt][i] + bias;
        }
    }
}

// ---------------------------------------------------------------------------
extern "C" void kernel_launch(void* const* d_in, const int* in_sizes, int n_in,
                              void* d_out, int out_size, void* d_ws, size_t ws_size,
                              hipStream_t stream) {
    const float*     H  = (const float*)d_in[0];
    const long long* ei = (const long long*)d_in[1];   // int64 edge_index
    const float*     EA = (const float*)d_in[2];
    const float*     We = (const float*)d_in[3];
    const float*     be = (const float*)d_in[4];
    const float*     W1 = (const float*)d_in[5];
    const float*     b1 = (const float*)d_in[6];
    const float*     W2 = (const float*)d_in[7];
    const float*     b2 = (const float*)d_in[8];
    float* out = (float*)d_out;
    float* agg = (float*)d_ws;                          // [N_NODES, IN_F] fp32 scratch

    // 1) zero the aggregation buffer
    const int n4 = N_NODES * IN_F / 4;
    gine_zero_kernel<<<(n4 + 255) / 256, 256, 0, stream>>>((float4*)agg, n4);

    // 2) edge pass: 16 edges per wave, 8 waves per block
    gine_edge_kernel<<<N_EDGES / (16 * 8), 256, 0, stream>>>(H, ei, EA, We, be, agg);

    // 3) fused node MLP: 16 nodes per single-wave block
    gine_node_kernel<<<N_NODES / 16, 32, 0, stream>>>(agg, W1, b1, W2, b2, out);
}